// ChebNet_53764400611981
// MI455X (gfx1250) — compile-verified
//
#include <hip/hip_runtime.h>
#include <hip/hip_bf16.h>

// ---------------------------------------------------------------------------
// ChebNet forward for MI455X (gfx1250, wave32).
//
// Strategy:
//  * Chebyshev recursion = 4-point grid stencil in LDS (L has <=4 nnz/row),
//    NOT a dense GEMM (saves ~0.93 TFLOP of wasted fp32 work).
//  * Per-point linear [Cin*5 -> Fout] = fp32 WMMA (v_wmma_f32_16x16x4_f32),
//    fused with the stencil so Cheb features never hit global memory.
//  * BN stats via atomic block-reductions, then fused BN+ReLU elementwise.
//  * Activations ping-pong in d_ws (2 x 100.7 MB), stats tail (1.5 KB).
// ---------------------------------------------------------------------------

#define V_NODES   1024
#define GRID_N    32
#define BATCH     128
#define KCH       5        // Chebyshev order
#define CC        8        // channel chunk held in LDS
#define FO_TILE   48       // output channels per workgroup (3 x 16)
#define NTHREADS  512
#define NWAVES    16       // 512 / wave32

typedef float v2f __attribute__((ext_vector_type(2)));
typedef float v8f __attribute__((ext_vector_type(8)));

// ---------------------------------------------------------------------------
// Fused Chebyshev-stencil + WMMA GEMM.
// Grid: (Fout/FO_TILE, BATCH). Block: 512 threads = 16 waves.
// Each WG produces hout[b, o0..o0+47, 0..1023] (pre-BN).
// ---------------------------------------------------------------------------
__global__ __launch_bounds__(NTHREADS) void cheb_gemm_kernel(
    const float* __restrict__ hin,   // [B, Cin, V]
    const float* __restrict__ Lmat,  // [V, V] dense rescaled Laplacian
    const float* __restrict__ W,     // [Fout, Cin*KCH]  (K fastest)
    float* __restrict__ hout,        // [B, Fout, V]
    int Cin, int Fout)
{
    __shared__ float Tbuf[3][V_NODES][CC + 1];       // 110.6 KB : T_{k-2},T_{k-1},T_k
    __shared__ float Wst[FO_TILE][CC * KCH + 1];     // 7.9 KB   : weight slice
    __shared__ float Lnb[V_NODES][4];                // 16 KB    : 4 nnz per row of L
    __shared__ float Ytr[NWAVES][16][17];            // 17.4 KB  : store-transpose

    const int t    = threadIdx.x;
    const int b    = blockIdx.y;
    const int o0   = blockIdx.x * FO_TILE;
    const int wave = t >> 5;
    const int lane = t & 31;
    const int l15  = lane & 15;
    const int hi   = lane >> 4;

    // ---- load the 4 neighbor Laplacian entries per node (grid graph) ------
    for (int u = t; u < V_NODES; u += NTHREADS) {
        const float* Lr = Lmat + (size_t)u * V_NODES;
        Lnb[u][0] = (u >= GRID_N)                  ? Lr[u - GRID_N] : 0.0f; // north
        Lnb[u][1] = ((u & (GRID_N - 1)) != 0)      ? Lr[u - 1]      : 0.0f; // west
        Lnb[u][2] = ((u & (GRID_N - 1)) != GRID_N - 1) ? Lr[u + 1]  : 0.0f; // east
        Lnb[u][3] = (u < V_NODES - GRID_N)         ? Lr[u + GRID_N] : 0.0f; // south
    }

    v8f acc[12];                                    // 4 v-tiles x 3 o-tiles
    #pragma unroll
    for (int i = 0; i < 12; ++i)
        #pragma unroll
        for (int j = 0; j < 8; ++j) acc[i][j] = 0.0f;

    for (int c0 = 0; c0 < Cin; c0 += CC) {
        // ---- T0 chunk = x[b, c0..c0+7, :] (coalesced over v) --------------
        for (int idx = t; idx < V_NODES * CC; idx += NTHREADS) {
            int u  = idx & (V_NODES - 1);
            int ci = idx >> 10;
            int c  = c0 + ci;
            Tbuf[0][u][ci] = (c < Cin) ? hin[((size_t)b * Cin + c) * V_NODES + u]
                                       : 0.0f;
        }
        // ---- stage weight slice: Wst[oo][ci*5+kk] -------------------------
        for (int idx = t; idx < FO_TILE * CC * KCH; idx += NTHREADS) {
            int oo = idx / (CC * KCH);
            int r  = idx % (CC * KCH);
            int ci = r / KCH;
            int c  = c0 + ci;
            Wst[oo][r] = (c < Cin)
                ? W[(size_t)(o0 + oo) * ((size_t)Cin * KCH) + (size_t)c * KCH + (r % KCH)]
                : 0.0f;
        }
        __syncthreads();

        int pm1 = 0, pc = 0;   // buffer indices of T_{k-1}, T_k
        for (int kk = 0; kk < KCH; ++kk) {
            if (kk >= 1) {
                int dst, src, sub;
                if (kk == 1) { dst = 1; src = 0; sub = 0; }       // T1 = L*T0
                else         { dst = (pc + 1) % 3; src = pc; sub = pm1; }
                for (int u = t; u < V_NODES; u += NTHREADS) {
                    float l0 = Lnb[u][0], l1 = Lnb[u][1];
                    float l2 = Lnb[u][2], l3 = Lnb[u][3];
                    int un = (u >= GRID_N) ? u - GRID_N : u;      // coeff 0 when clamped
                    int uw = (u > 0) ? u - 1 : u;
                    int ue = (u < V_NODES - 1) ? u + 1 : u;
                    int us = (u < V_NODES - GRID_N) ? u + GRID_N : u;
                    #pragma unroll
                    for (int ci = 0; ci < CC; ++ci) {
                        float s = l0 * Tbuf[src][un][ci] + l1 * Tbuf[src][uw][ci]
                                + l2 * Tbuf[src][ue][ci] + l3 * Tbuf[src][us][ci];
                        Tbuf[dst][u][ci] = (kk == 1) ? s : 2.0f * s - Tbuf[sub][u][ci];
                    }
                }
                if (kk == 1) { pm1 = 0; pc = 1; }
                else         { pm1 = pc; pc = dst; }
                __syncthreads();
            }

            // ---- WMMA accumulate: Y += T_kk_chunk x Wslice ----------------
            // fp32 A 16x4: VGPR0 = K (hi?2:0), VGPR1 = K (hi?3:1); M = lane&15.
            #pragma unroll
            for (int vt = 0; vt < 4; ++vt) {
                int vrow = (wave * 4 + vt) * 16 + l15;
                #pragma unroll
                for (int ks = 0; ks < CC; ks += 4) {
                    v2f A;
                    A.x = Tbuf[pc][vrow][ks + (hi ? 2 : 0)];
                    A.y = Tbuf[pc][vrow][ks + (hi ? 3 : 1)];
                    #pragma unroll
                    for (int ot = 0; ot < 3; ++ot) {
                        int oo = ot * 16 + l15;
                        v2f Bv;
                        Bv.x = Wst[oo][(ks + (hi ? 2 : 0)) * KCH + kk];
                        Bv.y = Wst[oo][(ks + (hi ? 3 : 1)) * KCH + kk];
                        acc[vt * 3 + ot] = __builtin_amdgcn_wmma_f32_16x16x4_f32(
                            false, A, false, Bv, (short)0, acc[vt * 3 + ot],
                            false, false);
                    }
                }
            }
            __syncthreads();
        }
    }

    // ---- write out: transpose each 16x16 tile through LDS for coalescing --
    // D layout: acc[r] @ lane l -> (M = r + (hi?8:0), N = l&15); M=v, N=o.
    for (int vt = 0; vt < 4; ++vt) {
        int vb = (wave * 4 + vt) * 16;
        for (int ot = 0; ot < 3; ++ot) {
            v8f a = acc[vt * 3 + ot];
            #pragma unroll
            for (int r = 0; r < 8; ++r)
                Ytr[wave][r + (hi ? 8 : 0)][l15] = a[r];
            __syncthreads();
            int obase = o0 + ot * 16;
            #pragma unroll
            for (int it = 0; it < 8; ++it) {
                int orow = it * 2 + hi;
                float val = Ytr[wave][l15][orow];
                hout[((size_t)b * Fout + obase + orow) * V_NODES + vb + l15] = val;
            }
            __syncthreads();
        }
    }
}

// ---------------------------------------------------------------------------
// BN statistics: stats[o] = sum, stats[Fout+o] = sum of squares (atomic).
// Grid: (Fout, 16). Block: 256.
// ---------------------------------------------------------------------------
__global__ void zero_stats_kernel(float* __restrict__ stats, int n)
{
    int i = blockIdx.x * blockDim.x + threadIdx.x;
    if (i < n) stats[i] = 0.0f;
}

__global__ void bn_stats_kernel(const float* __restrict__ h,
                                float* __restrict__ stats, int Fout)
{
    const int o     = blockIdx.x;
    const int slice = blockIdx.y;                  // 16 slices of B*V
    const int per   = (BATCH * V_NODES) / 16;      // 8192
    float s = 0.0f, sq = 0.0f;
    for (int i = threadIdx.x; i < per; i += blockDim.x) {
        int gi = slice * per + i;
        int b  = gi >> 10;
        int v  = gi & (V_NODES - 1);
        float x = h[((size_t)b * Fout + o) * V_NODES + v];
        s += x; sq += x * x;
    }
    __shared__ float rs[256], rq[256];
    rs[threadIdx.x] = s; rq[threadIdx.x] = sq;
    __syncthreads();
    for (int st = 128; st > 0; st >>= 1) {
        if (threadIdx.x < st) {
            rs[threadIdx.x] += rs[threadIdx.x + st];
            rq[threadIdx.x] += rq[threadIdx.x + st];
        }
        __syncthreads();
    }
    if (threadIdx.x == 0) {
        atomicAdd(&stats[o],        rs[0]);
        atomicAdd(&stats[Fout + o], rq[0]);
    }
}

// ---------------------------------------------------------------------------
// BN (training-mode batch stats) + ReLU, in place.
// ---------------------------------------------------------------------------
__global__ void bn_relu_kernel(float* __restrict__ h,
                               const float* __restrict__ stats,
                               const float* __restrict__ g,
                               const float* __restrict__ bb, int Fout)
{
    size_t idx   = (size_t)blockIdx.x * blockDim.x + threadIdx.x;
    size_t total = (size_t)BATCH * Fout * V_NODES;
    if (idx >= total) return;
    int o = (int)((idx >> 10) % (size_t)Fout);
    const float n   = (float)(BATCH * V_NODES);
    float m   = stats[o] / n;
    float var = stats[Fout + o] / n - m * m;
    float x   = (h[idx] - m) * rsqrtf(var + 1e-5f) * g[o] + bb[o];
    h[idx] = x > 0.0f ? x : 0.0f;
}

// ---------------------------------------------------------------------------
// Classifier: out[b,o] = h[b,:] . clf_w[o,:] + clf_b[o].  Grid: (10, 128).
// ---------------------------------------------------------------------------
__global__ void classifier_kernel(const float* __restrict__ h,
                                  const float* __restrict__ w,
                                  const float* __restrict__ bias,
                                  float* __restrict__ out)
{
    const int o = blockIdx.x;
    const int b = blockIdx.y;
    const int F = 96 * V_NODES;
    const float* hb = h + (size_t)b * F;
    const float* wo = w + (size_t)o * F;
    float s = 0.0f;
    for (int i = threadIdx.x; i < F; i += blockDim.x) s += hb[i] * wo[i];
    __shared__ float rs[256];
    rs[threadIdx.x] = s;
    __syncthreads();
    for (int st = 128; st > 0; st >>= 1) {
        if (threadIdx.x < st) rs[threadIdx.x] += rs[threadIdx.x + st];
        __syncthreads();
    }
    if (threadIdx.x == 0) out[b * 10 + o] = rs[0] + bias[o];
}

// ---------------------------------------------------------------------------
extern "C" void kernel_launch(void* const* d_in, const int* in_sizes, int n_in,
                              void* d_out, int out_size, void* d_ws, size_t ws_size,
                              hipStream_t stream)
{
    (void)in_sizes; (void)n_in; (void)out_size; (void)ws_size;
    const float* x    = (const float*)d_in[0];
    const float* Lmat = (const float*)d_in[1];
    // dict order: x, L, then (w_i, g_i, b_i) for i=0..6, then clf_w, clf_b
    static const int CH_IN [7] = {  3, 96, 96,  96, 192, 192, 192 };
    static const int CH_OUT[7] = { 96, 96, 96, 192, 192, 192,  96 };

    char*  ws       = (char*)d_ws;
    size_t actBytes = (size_t)BATCH * V_NODES * 192 * sizeof(float); // 100.7 MB
    float* hA    = (float*)ws;
    float* hB    = (float*)(ws + actBytes);
    float* stats = (float*)(ws + 2 * actBytes);                     // 2*192 floats

    const float* cur = x;
    float*       nxt = hA;
    for (int i = 0; i < 7; ++i) {
        const float* W  = (const float*)d_in[2 + 3 * i];
        const float* g  = (const float*)d_in[3 + 3 * i];
        const float* bb = (const float*)d_in[4 + 3 * i];
        const int Cin = CH_IN[i], Fout = CH_OUT[i];

        dim3 grid(Fout / FO_TILE, BATCH);
        cheb_gemm_kernel<<<grid, NTHREADS, 0, stream>>>(cur, Lmat, W, nxt, Cin, Fout);

        zero_stats_kernel<<<2, 256, 0, stream>>>(stats, 2 * Fout);
        bn_stats_kernel<<<dim3(Fout, 16), 256, 0, stream>>>(nxt, stats, Fout);

        size_t total = (size_t)BATCH * Fout * V_NODES;
        bn_relu_kernel<<<(unsigned)((total + 255) / 256), 256, 0, stream>>>(
            nxt, stats, g, bb, Fout);

        cur = nxt;
        nxt = (nxt == hA) ? hB : hA;
    }

    classifier_kernel<<<dim3(10, BATCH), 256, 0, stream>>>(
        cur, (const float*)d_in[23], (const float*)d_in[24], (float*)d_out);
}